// GIN_16475494547884
// MI455X (gfx1250) — compile-verified
//
#include <hip/hip_runtime.h>

// GIN forward for MI455X (gfx1250, wave32, WMMA).
// d = 128. fp32 WMMA (V_WMMA_F32_16X16X4_F32) keeps fp32 accuracy; the op is
// L2-bandwidth bound on edge aggregation (3 x 328 MB gathers + 246M f32
// atomics, all resident in the 192 MB L2), so GEMMs use fp32 WMMA rather than
// a lossy bf16 downcast.
//
// LDS rows are padded to 132 floats (528 B): row-start bank advances 4 banks
// per row, so the row-per-lane b64 fragment loads of WMMA A/B operands touch
// all 64 banks exactly once (conflict-free), while keeping 8/16-byte
// alignment for v2f/float4 accesses.

typedef float v2f __attribute__((ext_vector_type(2)));
typedef float v8f __attribute__((ext_vector_type(8)));

#define D 128
#define S 132          // padded LDS row stride (floats)
#define TILE_ROWS 128  // nodes per block tile
#define NTHREADS 256   // 8 waves of 32

// ---------------------------------------------------------------------------
// Transpose a batch of contiguous 128x128 matrices: dst[m][n][k] = src[m][k][n]
__global__ void transpose128_kernel(float* __restrict__ dst,
                                    const float* __restrict__ src, int total) {
  int i = blockIdx.x * blockDim.x + threadIdx.x;
  if (i >= total) return;
  int mat = i >> 14;
  int n = (i >> 7) & (D - 1);
  int k = i & (D - 1);
  dst[i] = src[(mat << 14) + (k << 7) + n];
}

// ---------------------------------------------------------------------------
// z[i] = (1 + eps[layer]) * h[i], float4-vectorized
__global__ void gin_scale_kernel(float* __restrict__ z, const float* __restrict__ h,
                                 const float* __restrict__ eps, int layer, int total4) {
  int i = blockIdx.x * blockDim.x + threadIdx.x;
  if (i >= total4) return;
  float s = 1.0f + eps[layer];
  float4 v = reinterpret_cast<const float4*>(h)[i];
  v.x *= s; v.y *= s; v.z *= s; v.w *= s;
  reinterpret_cast<float4*>(z)[i] = v;
}

// ---------------------------------------------------------------------------
// z[dst] += h[src] over all edges. One wave per edge: 512B coalesced gather
// of h[src], 4x global_atomic_add_f32 per lane into z[dst] (L2-resident).
__global__ void gin_aggregate_kernel(float* __restrict__ z, const float* __restrict__ h,
                                     const int* __restrict__ src, const int* __restrict__ dst,
                                     int E) {
  int t = blockIdx.x * blockDim.x + threadIdx.x;
  int e = t >> 5;
  if (e >= E) return;
  int q = (t & 31) << 2;
  int s = src[e];
  int d = dst[e];
  const float4 v = *reinterpret_cast<const float4*>(h + (size_t)s * D + q);
  float* o = z + (size_t)d * D + q;
  atomicAdd(o + 0, v.x);
  atomicAdd(o + 1, v.y);
  atomicAdd(o + 2, v.z);
  atomicAdd(o + 3, v.w);
}

// ---------------------------------------------------------------------------
// One wave accumulates a 32x64 output tile (2 row sub-tiles x 4 col sub-tiles,
// 8 v8f accumulators) of Zl[128xD] @ W (transposed Wt[n][k] in LDS), K = D.
// A fragment: lane m = lane&15 holds row m, K pair {koff, koff+1},
// koff = (lane>>4)*2 -> contiguous b64 from row-major padded Zl.
// B fragment: lane holds sub-tile column n = lane&15, same K pair -> b64 from
// the transposed padded weight tile. B is reused across both row sub-tiles:
// 6 b64 loads per 8 WMMAs.
__device__ __forceinline__ void wave_gemm_32x64(const float* __restrict__ Zl,
                                                const float* __restrict__ Wt,
                                                int rowBase, int colBase, int lane,
                                                v8f acc[8]) {
  const int m = lane & 15;
  const int koff = (lane >> 4) << 1;  // 0 or 2
  const float* a0p = Zl + (rowBase + m) * S + koff;
  const float* a1p = a0p + 16 * S;
  const float* bp  = Wt + (colBase + m) * S + koff;
#pragma unroll
  for (int k = 0; k < D; k += 4) {
    v2f a0 = *reinterpret_cast<const v2f*>(a0p + k);
    v2f a1 = *reinterpret_cast<const v2f*>(a1p + k);
#pragma unroll
    for (int nt = 0; nt < 4; ++nt) {
      v2f b = *reinterpret_cast<const v2f*>(bp + nt * 16 * S + k);
      acc[nt] = __builtin_amdgcn_wmma_f32_16x16x4_f32(
          false, a0, false, b, (short)0, acc[nt], false, false);
      acc[4 + nt] = __builtin_amdgcn_wmma_f32_16x16x4_f32(
          false, a1, false, b, (short)0, acc[4 + nt], false, false);
    }
  }
}

// Stage a pre-transposed 128x128 matrix (row-major) into padded LDS. Coalesced
// float4 global reads, b128 LDS stores.
__device__ __forceinline__ void stage_w(float* __restrict__ WtL,
                                        const float* __restrict__ Wg, int tid) {
  for (int i = tid; i < (D * D) >> 2; i += NTHREADS) {
    int r = i >> 5;            // 32 float4 per row
    int c4 = (i & 31) << 2;
    *reinterpret_cast<float4*>(WtL + r * S + c4) =
        reinterpret_cast<const float4*>(Wg)[i];
  }
}

// Stage a TILE_ROWS x D activation tile into padded LDS (rows clamped at N-1).
__device__ __forceinline__ void stage_z(float* __restrict__ Zl,
                                        const float* __restrict__ zg,
                                        int nodeBase, int N, int tid) {
  for (int i = tid; i < (TILE_ROWS * D) >> 2; i += NTHREADS) {
    int r = i >> 5;
    int c4 = (i & 31) << 2;
    int node = nodeBase + r;
    if (node >= N) node = N - 1;  // clamp tail; stores are guarded later
    *reinterpret_cast<float4*>(Zl + r * S + c4) =
        *reinterpret_cast<const float4*>(zg + (size_t)node * D + c4);
  }
}

// ---------------------------------------------------------------------------
// Fused 2-layer MLP for one 128-node tile:
//   h_out = relu(relu(z @ W1 + b1) @ W2 + b2)
// LDS: Wt (128 x S, 66KB) + Zl (128 x S, 66KB). Weights pre-transposed in ws.
__global__ void __launch_bounds__(NTHREADS, 1)
gin_mlp_kernel(const float* __restrict__ z,
               const float* __restrict__ W1t, const float* __restrict__ b1,
               const float* __restrict__ W2t, const float* __restrict__ b2,
               float* __restrict__ hout, int N) {
  extern __shared__ float smem[];
  float* Wt = smem;        // 128 * S
  float* Zl = smem + D * S;

  const int tid = threadIdx.x;
  const int lane = tid & 31;
  const int wave = tid >> 5;
  const int wr = wave >> 1;  // 0..3: 32-row group
  const int wc = wave & 1;   // 0..1: 64-col group
  const int nodeBase = blockIdx.x * TILE_ROWS;

  stage_w(Wt, W1t, tid);
  stage_z(Zl, z, nodeBase, N, tid);
  __syncthreads();

  // GEMM 1
  v8f acc[8] = {};
  wave_gemm_32x64(Zl, Wt, wr * 32, wc * 64, lane, acc);
  __syncthreads();  // all fragment reads done before Zl/Wt are overwritten

  // bias + relu, write z2 back over the Zl buffer
  const int m = lane & 15;
  const int rlo = (lane >> 4) << 3;  // lanes 16-31 hold rows M+8
#pragma unroll
  for (int rt = 0; rt < 2; ++rt) {
#pragma unroll
    for (int nt = 0; nt < 4; ++nt) {
      int col = wc * 64 + nt * 16 + m;
      float bias = b1[col];
#pragma unroll
      for (int r = 0; r < 8; ++r) {
        float v = acc[rt * 4 + nt][r] + bias;
        Zl[(wr * 32 + rt * 16 + rlo + r) * S + col] = v > 0.0f ? v : 0.0f;
      }
    }
  }
  stage_w(Wt, W2t, tid);
  __syncthreads();

  // GEMM 2
  v8f acc2[8] = {};
  wave_gemm_32x64(Zl, Wt, wr * 32, wc * 64, lane, acc2);

  // bias + relu, store h tile
#pragma unroll
  for (int rt = 0; rt < 2; ++rt) {
#pragma unroll
    for (int nt = 0; nt < 4; ++nt) {
      int col = wc * 64 + nt * 16 + m;
      float bias = b2[col];
#pragma unroll
      for (int r = 0; r < 8; ++r) {
        int row = nodeBase + wr * 32 + rt * 16 + rlo + r;
        if (row < N) {
          float v = acc2[rt * 4 + nt][r] + bias;
          hout[(size_t)row * D + col] = v > 0.0f ? v : 0.0f;
        }
      }
    }
  }
}

// ---------------------------------------------------------------------------
// out = concat(x, h1, h2, h3) @ final_w + final_b
// K = 512 as 4 segments of 128 (FWt pre-transposed per segment in ws);
// accumulators persist across segments.
__global__ void __launch_bounds__(NTHREADS, 1)
gin_final_kernel(const float* __restrict__ x0, const float* __restrict__ x1,
                 const float* __restrict__ x2, const float* __restrict__ x3,
                 const float* __restrict__ FWt, const float* __restrict__ FB,
                 float* __restrict__ out, int N) {
  extern __shared__ float smem[];
  float* Wt = smem;
  float* Zl = smem + D * S;

  const int tid = threadIdx.x;
  const int lane = tid & 31;
  const int wave = tid >> 5;
  const int wr = wave >> 1;
  const int wc = wave & 1;
  const int nodeBase = blockIdx.x * TILE_ROWS;

  const float* xs[4] = {x0, x1, x2, x3};
  v8f acc[8] = {};

  for (int s = 0; s < 4; ++s) {
    stage_w(Wt, FWt + (size_t)s * D * D, tid);
    stage_z(Zl, xs[s], nodeBase, N, tid);
    __syncthreads();
    wave_gemm_32x64(Zl, Wt, wr * 32, wc * 64, lane, acc);
    __syncthreads();
  }

  const int m = lane & 15;
  const int rlo = (lane >> 4) << 3;
#pragma unroll
  for (int rt = 0; rt < 2; ++rt) {
#pragma unroll
    for (int nt = 0; nt < 4; ++nt) {
      int col = wc * 64 + nt * 16 + m;
      float bias = FB[col];
#pragma unroll
      for (int r = 0; r < 8; ++r) {
        int row = nodeBase + wr * 32 + rt * 16 + rlo + r;
        if (row < N) out[(size_t)row * D + col] = acc[rt * 4 + nt][r] + bias;
      }
    }
  }
}

// ---------------------------------------------------------------------------
extern "C" void kernel_launch(void* const* d_in, const int* in_sizes, int n_in,
                              void* d_out, int out_size, void* d_ws, size_t ws_size,
                              hipStream_t stream) {
  const float* x   = (const float*)d_in[0];
  const int*  edge = (const int*)d_in[1];   // [2, E] flat: src then dst
  const float* W1  = (const float*)d_in[2];
  const float* b1  = (const float*)d_in[3];
  const float* W2  = (const float*)d_in[4];
  const float* b2  = (const float*)d_in[5];
  const float* eps = (const float*)d_in[6];
  const float* FW  = (const float*)d_in[7];
  const float* FB  = (const float*)d_in[8];

  const int N = in_sizes[0] / D;
  const int E = in_sizes[1] / 2;
  const int L = in_sizes[2] / (D * D);        // = 3
  const int SEGS = in_sizes[7] / (D * D);     // = L + 1 = 4

  // Workspace layout (floats)
  float* zbuf = (float*)d_ws;                           // N*D accumulator
  float* hbuf = zbuf + (size_t)N * D;                   // L buffers of N*D
  float* w1t  = hbuf + (size_t)L * N * D;               // L * 128*128
  float* w2t  = w1t + (size_t)L * D * D;                // L * 128*128
  float* fwt  = w2t + (size_t)L * D * D;                // SEGS * 128*128

  const int total4 = (N * D) >> 2;
  const int tileBlocks = (N + TILE_ROWS - 1) / TILE_ROWS;
  const size_t shmem = (size_t)2 * D * S * sizeof(float);  // ~132 KB
  dim3 blk(NTHREADS);

  // One-time (per launch) weight transposes: tiny, keeps per-block staging
  // fully coalesced.
  transpose128_kernel<<<(L * D * D + NTHREADS - 1) / NTHREADS, blk, 0, stream>>>(
      w1t, W1, L * D * D);
  transpose128_kernel<<<(L * D * D + NTHREADS - 1) / NTHREADS, blk, 0, stream>>>(
      w2t, W2, L * D * D);
  transpose128_kernel<<<(SEGS * D * D + NTHREADS - 1) / NTHREADS, blk, 0, stream>>>(
      fwt, FW, SEGS * D * D);

  const float* hprev = x;
  for (int l = 0; l < L; ++l) {
    float* hl = hbuf + (size_t)l * N * D;
    gin_scale_kernel<<<(total4 + NTHREADS - 1) / NTHREADS, blk, 0, stream>>>(
        zbuf, hprev, eps, l, total4);
    gin_aggregate_kernel<<<(E * 32 + NTHREADS - 1) / NTHREADS, blk, 0, stream>>>(
        zbuf, hprev, edge, edge + E, E);
    gin_mlp_kernel<<<tileBlocks, blk, shmem, stream>>>(
        zbuf, w1t + (size_t)l * D * D, b1 + (size_t)l * D,
        w2t + (size_t)l * D * D, b2 + (size_t)l * D, hl, N);
    hprev = hl;
  }
  gin_final_kernel<<<tileBlocks, blk, shmem, stream>>>(
      x, hbuf, hbuf + (size_t)N * D, hbuf + 2 * (size_t)N * D,
      fwt, FB, (float*)d_out, N);
}